// CascadeAttention_64132451664401
// MI455X (gfx1250) — compile-verified
//
#include <hip/hip_runtime.h>
#include <math.h>

typedef __attribute__((ext_vector_type(2))) float v2f;
typedef __attribute__((ext_vector_type(8))) float v8f;

#define NUM_HEADS     32
#define NUM_KV_HEADS  8
#define HEAD_DIM      128
#define PAGE_SIZE     16
#define GQA           4                       // NUM_HEADS / NUM_KV_HEADS
#define ATT_SCALE     0.08838834764831845f
#define WAVES         4
#define TOK_STRIDE    (NUM_KV_HEADS * HEAD_DIM)   // 1024 floats between tokens

__device__ __forceinline__ float half16_max(float v) {
  v = fmaxf(v, __shfl_xor(v, 1, 32));
  v = fmaxf(v, __shfl_xor(v, 2, 32));
  v = fmaxf(v, __shfl_xor(v, 4, 32));
  v = fmaxf(v, __shfl_xor(v, 8, 32));
  return v;
}
__device__ __forceinline__ float half16_sum(float v) {
  v += __shfl_xor(v, 1, 32);
  v += __shfl_xor(v, 2, 32);
  v += __shfl_xor(v, 4, 32);
  v += __shfl_xor(v, 8, 32);
  return v;
}

// One workgroup per (kv_head, batch). 4 waves split the 352 pages (split-K),
// each wave runs an fp32 flash-attention pass with V_WMMA_F32_16X16X4_F32,
// partial states are LSE-merged through LDS at the end.
__global__ __launch_bounds__(32 * WAVES)
void cascade_attn_kernel(const float* __restrict__ q,
                         const float* __restrict__ kv_cache,
                         const int*  __restrict__ shared_page_idx,
                         const int*  __restrict__ seq1_page_idx,
                         const int*  __restrict__ seq2_page_idx,
                         float* __restrict__ out,
                         int l0_pages, int l1_pages, int l2_pages)
{
  __shared__ float lds_S[WAVES][16 * 16];            // per-wave S->P transpose bounce
  __shared__ float lds_O[WAVES][GQA][HEAD_DIM];      // per-wave partial outputs
  __shared__ float lds_M[WAVES][GQA];
  __shared__ float lds_L[WAVES][GQA];

  const int kh   = blockIdx.x;
  const int b    = blockIdx.y;
  const int tid  = threadIdx.x;
  const int lane = tid & 31;
  const int wave = tid >> 5;
  const int row  = lane & 15;   // M/N index within 16-lane half
  const int half = lane >> 4;   // selects K-pair 0-1 vs 2-3 in A/B operands

  const int total_tiles = l0_pages + l1_pages + l2_pages;
  const size_t page_stride = (size_t)2 * PAGE_SIZE * NUM_KV_HEADS * HEAD_DIM; // floats

  // ---- Q as resident WMMA A-operands: 32 k-steps of 16x4 over D=128.
  // Rows 0..3 = the GQA group; rows 4..15 zero-padded.
  const float* qb = q + (size_t)b * (NUM_HEADS * HEAD_DIM) + (size_t)kh * (GQA * HEAD_DIM);
  v2f qa[32];
  #pragma unroll
  for (int j = 0; j < 32; ++j) {
    const int d0 = 4 * j + 2 * half;
    float a0 = 0.0f, a1 = 0.0f;
    if (row < GQA) { a0 = qb[row * HEAD_DIM + d0]; a1 = qb[row * HEAD_DIM + d0 + 1]; }
    qa[j].x = a0; qa[j].y = a1;
  }

  // ---- flash state (rows 0..3, replicated across lanes 0..15 via butterflies)
  float m_s[GQA], l_s[GQA];
  #pragma unroll
  for (int g = 0; g < GQA; ++g) { m_s[g] = -1e30f; l_s[g] = 0.0f; }
  v8f acc[8];                                        // O accumulators, 8 d-blocks of 16
  #pragma unroll
  for (int blk = 0; blk < 8; ++blk)
    #pragma unroll
    for (int g = 0; g < 8; ++g) acc[blk][g] = 0.0f;

  for (int t = wave; t < total_tiles; t += WAVES) {
    // ---- page lookup across the three cascade levels
    int page;
    if (t < l0_pages)                 page = shared_page_idx[t];
    else if (t < l0_pages + l1_pages) page = seq1_page_idx[b * l1_pages + (t - l0_pages)];
    else                              page = seq2_page_idx[b * l2_pages + (t - l0_pages - l1_pages)];

    const float* kbase = kv_cache + (size_t)page * page_stride + (size_t)kh * HEAD_DIM;
    const float* vbase = kbase + (size_t)PAGE_SIZE * TOK_STRIDE;

    // ---- prefetch next tile's K/V pages (global_prefetch_b8)
    const int t2 = t + WAVES;
    if (t2 < total_tiles) {
      int p2;
      if (t2 < l0_pages)                 p2 = shared_page_idx[t2];
      else if (t2 < l0_pages + l1_pages) p2 = seq1_page_idx[b * l1_pages + (t2 - l0_pages)];
      else                               p2 = seq2_page_idx[b * l2_pages + (t2 - l0_pages - l1_pages)];
      const float* nk = kv_cache + (size_t)p2 * page_stride + (size_t)kh * HEAD_DIM;
      __builtin_prefetch(nk + (size_t)row * TOK_STRIDE + half * 64, 0, 0);                            // K
      __builtin_prefetch(nk + (size_t)PAGE_SIZE * TOK_STRIDE + (size_t)row * TOK_STRIDE + half * 64, 0, 0); // V
    }

    // ---- S = Q K^T : 32 chained V_WMMA_F32_16X16X4_F32.
    // B operand: lane=token n, VGPR pair = K[n, d0+2h], K[n, d0+2h+1]  (float2 loads)
    const float* krow = kbase + (size_t)row * TOK_STRIDE + 2 * half;
    v8f s;
    #pragma unroll
    for (int g = 0; g < 8; ++g) s[g] = 0.0f;
    #pragma unroll
    for (int j = 0; j < 32; ++j) {
      v2f kb = *(const v2f*)(krow + 4 * j);
      s = __builtin_amdgcn_wmma_f32_16x16x4_f32(false, qa[j], false, kb,
                                                (short)0, s, false, false);
    }

    // ---- online softmax on the 4 real rows (C-layout: row g lives in s[g], lanes 0..15)
    float pr[8];
    #pragma unroll
    for (int g = 0; g < GQA; ++g) {
      const float sv = s[g] * ATT_SCALE;
      const float mn = fmaxf(m_s[g], half16_max(sv));
      const float e  = half ? 0.0f : __expf(sv - mn);   // zero padded rows 8..15
      const float rs = half16_sum(e);
      const float corr = __expf(m_s[g] - mn);
      l_s[g] = l_s[g] * corr + rs;
      m_s[g] = mn;
      pr[g] = e;
      #pragma unroll
      for (int blk = 0; blk < 8; ++blk) acc[blk][g] *= corr;  // rescale O row g
    }
    #pragma unroll
    for (int g = GQA; g < 8; ++g) pr[g] = 0.0f;               // zero padded rows 4..7

    // ---- transpose P (C-layout -> A-layout) through per-wave LDS tile
    float* Sl = lds_S[wave];
    #pragma unroll
    for (int r = 0; r < 8; ++r)
      Sl[(r + 8 * half) * 16 + row] = pr[r];
    __asm__ volatile("s_wait_dscnt 0" ::: "memory");          // same-wave ds RAW
    v2f pa[4];
    #pragma unroll
    for (int i = 0; i < 4; ++i)
      pa[i] = *(const v2f*)(Sl + row * 16 + 4 * i + 2 * half);

    // ---- O += P V : 8 d-blocks x 4 k-steps of V_WMMA_F32_16X16X4_F32.
    // B operand: lane = output col d within block, VGPR pair = V[tok, d], V[tok+1, d]
    #pragma unroll
    for (int blk = 0; blk < 8; ++blk) {
      #pragma unroll
      for (int i = 0; i < 4; ++i) {
        const int tok = 4 * i + 2 * half;
        const float* vp = vbase + (size_t)tok * TOK_STRIDE + blk * 16 + row;
        v2f vb;
        vb.x = vp[0];
        vb.y = vp[TOK_STRIDE];
        acc[blk] = __builtin_amdgcn_wmma_f32_16x16x4_f32(false, pa[i], false, vb,
                                                         (short)0, acc[blk], false, false);
      }
    }
  }

  // ---- publish per-wave partial state (rows 0..3 live in lanes 0..15)
  if (half == 0) {
    #pragma unroll
    for (int blk = 0; blk < 8; ++blk)
      #pragma unroll
      for (int g = 0; g < GQA; ++g)
        lds_O[wave][g][blk * 16 + row] = acc[blk][g];
    if (row == 0) {
      #pragma unroll
      for (int g = 0; g < GQA; ++g) { lds_M[wave][g] = m_s[g]; lds_L[wave][g] = l_s[g]; }
    }
  }
  __syncthreads();

  // ---- LSE merge of the 4 wave-partials; 128 threads cover d, loop over g
  float* ob = out + (size_t)b * (NUM_HEADS * HEAD_DIM) + (size_t)kh * (GQA * HEAD_DIM);
  #pragma unroll
  for (int g = 0; g < GQA; ++g) {
    float M = -1e30f;
    #pragma unroll
    for (int w = 0; w < WAVES; ++w) M = fmaxf(M, lds_M[w][g]);
    float L = 0.0f, O = 0.0f;
    #pragma unroll
    for (int w = 0; w < WAVES; ++w) {
      const float f = __expf(lds_M[w][g] - M);
      L += lds_L[w][g] * f;
      O += lds_O[w][g][tid] * f;
    }
    ob[g * HEAD_DIM + tid] = O / L;
  }
}

extern "C" void kernel_launch(void* const* d_in, const int* in_sizes, int n_in,
                              void* d_out, int out_size, void* d_ws, size_t ws_size,
                              hipStream_t stream) {
  (void)n_in; (void)out_size; (void)d_ws; (void)ws_size;
  const float* q  = (const float*)d_in[0];
  const float* kv = (const float*)d_in[1];
  const int* sp   = (const int*)d_in[2];
  const int* s1   = (const int*)d_in[3];
  const int* s2   = (const int*)d_in[4];
  float* out      = (float*)d_out;

  const int B  = in_sizes[0] / (NUM_HEADS * HEAD_DIM);
  const int l0 = in_sizes[2];
  const int l1 = in_sizes[3] / B;
  const int l2 = in_sizes[4] / B;

  dim3 grid(NUM_KV_HEADS, B);
  cascade_attn_kernel<<<grid, 32 * WAVES, 0, stream>>>(q, kv, sp, s1, s2, out, l0, l1, l2);
}